// NaiveBranchedLinear_40175124086889
// MI455X (gfx1250) — compile-verified
//
#include <hip/hip_runtime.h>

typedef __attribute__((ext_vector_type(2))) float v2f;
typedef __attribute__((ext_vector_type(8))) float v8f;

#define BATCH        1024
#define INPUT_DIM    4096
#define SOMA_UNITS   1024
#define BRANCHES     16
#define SAMPLES      16
#define NUM_CLASSES  1000
#define NEG_SLOPE    0.1f

#define ROWS 8  // batch rows staged in LDS per workgroup (8 * 4096 * 4B = 128 KB < 320 KB WGP LDS)

// ---------------------------------------------------------------------------
// Kernel 1: gather + branch linear + cable reduction + leaky ReLU -> soma
// One block = 256 threads handles ROWS batch rows for ALL soma units.
// Each thread owns one soma unit (4 iterations of 256 units) and accumulates
// 16 branches x 16 samples out of LDS for all ROWS rows.
// ---------------------------------------------------------------------------
__global__ __launch_bounds__(256) void dendrite_soma_kernel(
    const float* __restrict__ x,      // [BATCH][INPUT_DIM]
    const int*   __restrict__ idx,    // [U*BR][SAMPLES]
    const float* __restrict__ sw,     // [U*BR][SAMPLES]
    const float* __restrict__ sb,     // [U*BR]
    const float* __restrict__ cw,     // [U][BR]
    const float* __restrict__ somab,  // [U]
    float*       __restrict__ soma_out) // [BATCH][U]
{
    extern __shared__ float lds[];    // ROWS * INPUT_DIM floats
    const int tid     = threadIdx.x;
    const int rowBase = blockIdx.x * ROWS;

    // Stage ROWS rows of x into LDS with float4 vector loads.
    const float4* x4 = (const float4*)(x + (size_t)rowBase * INPUT_DIM);
    float4* l4 = (float4*)lds;
    #pragma unroll
    for (int i = 0; i < (ROWS * INPUT_DIM / 4) / 256; ++i)
        l4[tid + i * 256] = x4[tid + i * 256];
    __syncthreads();

    for (int uu = 0; uu < SOMA_UNITS / 256; ++uu) {
        const int u = uu * 256 + tid;
        float acc[ROWS];
        #pragma unroll
        for (int r = 0; r < ROWS; ++r) acc[r] = 0.f;

        #pragma unroll 1
        for (int br = 0; br < BRANCHES; ++br) {
            const int d = u * BRANCHES + br;
            const int4*   di = (const int4*)  (idx + (size_t)d * SAMPLES);
            const float4* dw = (const float4*)(sw  + (size_t)d * SAMPLES);
            const float bias = sb[d];
            float bl[ROWS];
            #pragma unroll
            for (int r = 0; r < ROWS; ++r) bl[r] = bias;

            #pragma unroll
            for (int q = 0; q < SAMPLES / 4; ++q) {
                const int4   iv = di[q];
                const float4 wv = dw[q];
                #pragma unroll
                for (int r = 0; r < ROWS; ++r) {
                    bl[r] += lds[r * INPUT_DIM + iv.x] * wv.x;
                    bl[r] += lds[r * INPUT_DIM + iv.y] * wv.y;
                    bl[r] += lds[r * INPUT_DIM + iv.z] * wv.z;
                    bl[r] += lds[r * INPUT_DIM + iv.w] * wv.w;
                }
            }
            const float c = cw[(size_t)u * BRANCHES + br];
            #pragma unroll
            for (int r = 0; r < ROWS; ++r) acc[r] += c * bl[r];
        }

        const float sbias = somab[u];
        #pragma unroll
        for (int r = 0; r < ROWS; ++r) {
            float v = acc[r] + sbias;
            v = (v >= 0.f) ? v : NEG_SLOPE * v;           // leaky relu
            soma_out[(size_t)(rowBase + r) * SOMA_UNITS + u] = v;
        }
    }
}

// ---------------------------------------------------------------------------
// Kernel 2: out = soma @ clsW^T + clsB using native fp32 WMMA 16x16x4.
// Block: 256 thr = 8 waves; wave tile 16(M)x64(N) (4 v8f accumulators,
// A fragment reused). Block tile 64(M)x128(N). Grid: (8 N-tiles, 16 M-tiles).
// Fragment mapping (ISA 7.12.2, 32-bit A 16x4 / C 16x16):
//   A: m = lane&15, k = 2*(lane>>4)+{0,1}  -> float2 from soma[m][k]
//   B: n = lane&15, k = 2*(lane>>4)+{0,1}  -> float2 from W[n][k]   (B = W^T)
//   D: vgpr v -> m = 8*(lane>>4)+v, n = lane&15
// ---------------------------------------------------------------------------
__global__ __launch_bounds__(256) void classifier_wmma_kernel(
    const float* __restrict__ soma,   // [BATCH][SOMA_UNITS]
    const float* __restrict__ clsW,   // [NUM_CLASSES][SOMA_UNITS]
    const float* __restrict__ clsB,   // [NUM_CLASSES]
    float*       __restrict__ out)    // [BATCH][NUM_CLASSES]
{
    const int lane     = threadIdx.x & 31;
    const int wave     = threadIdx.x >> 5;
    const int wm       = wave & 3;          // 4 waves along M
    const int wn       = wave >> 2;         // 2 waves along N
    const int lane15   = lane & 15;
    const int laneHalf = lane >> 4;         // 0 or 1
    const int kHi      = laneHalf * 2;      // K sub-offset within fragment

    const int mBase = blockIdx.y * 64 + wm * 16;
    const int nBase = blockIdx.x * 128 + wn * 64;

    const float* aRow = soma + (size_t)(mBase + lane15) * SOMA_UNITS + kHi;

    const float* bRow[4];
    #pragma unroll
    for (int t = 0; t < 4; ++t) {
        int n  = nBase + t * 16 + lane15;
        int nc = (n < NUM_CLASSES) ? n : 0;   // clamp OOB lanes to a valid row
        bRow[t] = clsW + (size_t)nc * SOMA_UNITS + kHi;
    }

    v8f acc0 = {}, acc1 = {}, acc2 = {}, acc3 = {};

    #pragma unroll 4
    for (int k = 0; k < SOMA_UNITS; k += 4) {
        v2f a  = *(const v2f*)(aRow + k);
        v2f b0 = *(const v2f*)(bRow[0] + k);
        v2f b1 = *(const v2f*)(bRow[1] + k);
        v2f b2 = *(const v2f*)(bRow[2] + k);
        v2f b3 = *(const v2f*)(bRow[3] + k);
        acc0 = __builtin_amdgcn_wmma_f32_16x16x4_f32(false, a, false, b0, (short)0, acc0, false, false);
        acc1 = __builtin_amdgcn_wmma_f32_16x16x4_f32(false, a, false, b1, (short)0, acc1, false, false);
        acc2 = __builtin_amdgcn_wmma_f32_16x16x4_f32(false, a, false, b2, (short)0, acc2, false, false);
        acc3 = __builtin_amdgcn_wmma_f32_16x16x4_f32(false, a, false, b3, (short)0, acc3, false, false);
    }

    const int mRow = mBase + laneHalf * 8;
    v8f accs[4] = {acc0, acc1, acc2, acc3};
    #pragma unroll
    for (int t = 0; t < 4; ++t) {
        const int n = nBase + t * 16 + lane15;
        if (n < NUM_CLASSES) {
            const float bias = clsB[n];
            #pragma unroll
            for (int v = 0; v < 8; ++v)
                out[(size_t)(mRow + v) * NUM_CLASSES + n] = accs[t][v] + bias;
        }
    }
}

// ---------------------------------------------------------------------------
extern "C" void kernel_launch(void* const* d_in, const int* in_sizes, int n_in,
                              void* d_out, int out_size, void* d_ws, size_t ws_size,
                              hipStream_t stream) {
    const float* x     = (const float*)d_in[0];
    const int*   idx   = (const int*)  d_in[1];
    const float* sw    = (const float*)d_in[2];
    const float* sb    = (const float*)d_in[3];
    const float* cw    = (const float*)d_in[4];
    const float* somab = (const float*)d_in[5];
    const float* clsW  = (const float*)d_in[6];
    const float* clsB  = (const float*)d_in[7];
    float* out  = (float*)d_out;
    float* soma = (float*)d_ws;   // 1024*1024 fp32 = 4 MB scratch

    const size_t ldsBytes = (size_t)ROWS * INPUT_DIM * sizeof(float);  // 128 KB
    dendrite_soma_kernel<<<BATCH / ROWS, 256, ldsBytes, stream>>>(
        x, idx, sw, sb, cw, somab, soma);

    dim3 g2(8, 16);  // N tiles (128 wide) x M tiles (64 tall)
    classifier_wmma_kernel<<<g2, 256, 0, stream>>>(soma, clsW, clsB, out);
}